// GraphModel_9964324127379
// MI455X (gfx1250) — compile-verified
//
#include <hip/hip_runtime.h>
#include <hip/hip_bf16.h>

typedef __attribute__((ext_vector_type(2))) float v2f;
typedef __attribute__((ext_vector_type(8))) float v8f;

#define G_ 64
#define M_ 256
#define N_ 256
#define DN_ 198
#define DNP_ 200   // padded to multiple of 4 for K-loop
#define DA_ 32

// ---------------------------------------------------------------------------
// Kernel 1: row-normalize (cosine prep) + transpose to [g][d][row] layout so
// WMMA A/B operand loads are contiguous across lanes. Pads K to Dpad with 0.
// One wave per row.
// ---------------------------------------------------------------------------
__global__ void normalize_t_kernel(const float* __restrict__ src,
                                   float* __restrict__ dst,
                                   int rows, int D, int Dpad) {
    int wid  = (blockIdx.x * blockDim.x + threadIdx.x) >> 5;
    int lane = threadIdx.x & 31;
    int g = wid / rows;
    int r = wid - g * rows;
    const float* s = src + ((size_t)g * rows + r) * D;
    float ss = 0.f;
    for (int d = lane; d < D; d += 32) { float v = s[d]; ss += v * v; }
    #pragma unroll
    for (int off = 16; off; off >>= 1) ss += __shfl_xor(ss, off, 32);
    float scale = rsqrtf(fmaxf(ss, 1e-12f));
    float* dbase = dst + (size_t)g * Dpad * rows + r;
    for (int d = lane; d < Dpad; d += 32) {
        float v = (d < D) ? s[d] * scale : 0.f;
        dbase[(size_t)d * rows] = v;
    }
}

// ---------------------------------------------------------------------------
// Kernel 2: per-(g, 16x16 tile): node_sim & app_sim via V_WMMA_F32_16X16X4_F32,
// hIoU + tiny MLP in VALU on the WMMA C-layout, store corr[g][m][n].
// Block = 256 threads = 8 waves; wave w handles n-tile (blockIdx.y*8 + w).
// ---------------------------------------------------------------------------
__global__ __launch_bounds__(256, 1)
void corr_kernel(const float* __restrict__ traTn,  // [G][200][256]
                 const float* __restrict__ detTn,  // [G][200][256]
                 const float* __restrict__ traTa,  // [G][32][256]
                 const float* __restrict__ detTa,  // [G][32][256]
                 const float* __restrict__ tra_box, // [G][256][4]
                 const float* __restrict__ det_box, // [G][256][4]
                 const float* __restrict__ W1, const float* __restrict__ b1,
                 const float* __restrict__ W2, const float* __restrict__ b2,
                 const float* __restrict__ W3, const float* __restrict__ b3,
                 float* __restrict__ corr)          // [G][256][256]
{
    __shared__ float sw[353]; // W1(48) b1(16) W2(256) b2(16) W3(16) b3(1)
    for (int i = threadIdx.x; i < 353; i += blockDim.x) {
        float v;
        if      (i < 48)  v = W1[i];
        else if (i < 64)  v = b1[i - 48];
        else if (i < 320) v = W2[i - 64];
        else if (i < 336) v = b2[i - 320];
        else if (i < 352) v = W3[i - 336];
        else              v = b3[0];
        sw[i] = v;
    }
    __syncthreads();
    const float* sW1 = sw;
    const float* sb1 = sw + 48;
    const float* sW2 = sw + 64;
    const float* sb2 = sw + 320;
    const float* sW3 = sw + 336;
    const float  sb3 = sw[352];

    int lane  = threadIdx.x & 31;
    int wave  = threadIdx.x >> 5;
    int g     = blockIdx.z;
    int mbase = blockIdx.x * 16;
    int nbase = (blockIdx.y * 8 + wave) * 16;
    int idx16 = lane & 15;
    int khalf = (lane >> 4) * 2;   // lanes 0-15 hold K={k,k+1}, lanes 16-31 K={k+2,k+3}

    // node cosine GEMM, K = 200 (padded)
    v8f accN = {};
    {
        const float* tA = traTn + (size_t)g * DNP_ * M_ + mbase + idx16;
        const float* tB = detTn + (size_t)g * DNP_ * N_ + nbase + idx16;
        for (int k0 = 0; k0 < DNP_; k0 += 4) {
            v2f a, b;
            a.x = tA[(size_t)(k0 + khalf)     * M_];
            a.y = tA[(size_t)(k0 + khalf + 1) * M_];
            b.x = tB[(size_t)(k0 + khalf)     * N_];
            b.y = tB[(size_t)(k0 + khalf + 1) * N_];
            accN = __builtin_amdgcn_wmma_f32_16x16x4_f32(
                false, a, false, b, (short)0, accN, false, false);
        }
    }
    // appearance cosine GEMM, K = 32
    v8f accA = {};
    {
        const float* tA = traTa + (size_t)g * DA_ * M_ + mbase + idx16;
        const float* tB = detTa + (size_t)g * DA_ * N_ + nbase + idx16;
        for (int k0 = 0; k0 < DA_; k0 += 4) {
            v2f a, b;
            a.x = tA[(size_t)(k0 + khalf)     * M_];
            a.y = tA[(size_t)(k0 + khalf + 1) * M_];
            b.x = tB[(size_t)(k0 + khalf)     * N_];
            b.y = tB[(size_t)(k0 + khalf + 1) * N_];
            accA = __builtin_amdgcn_wmma_f32_16x16x4_f32(
                false, a, false, b, (short)0, accA, false, false);
        }
    }

    // C/D layout: VGPR i, lanes 0-15 -> M=i, N=lane; lanes 16-31 -> M=i+8, N=lane-16
    int n = nbase + idx16;
    const float4* db4 = (const float4*)(det_box + (size_t)g * N_ * 4);
    const float4* tb4 = (const float4*)(tra_box + (size_t)g * M_ * 4);
    float4 B = db4[n];
    float hb = B.w - B.y;
    float area_b = (B.z - B.x) * hb;
    int moff = 8 * (lane >> 4);

    for (int i = 0; i < 8; ++i) {
        int m = mbase + i + moff;
        float4 A = tb4[m];
        float iw = fminf(A.z, B.z) - fmaxf(A.x, B.x); iw = fmaxf(iw, 0.f);
        float ih = fminf(A.w, B.w) - fmaxf(A.y, B.y); ih = fmaxf(ih, 0.f);
        float inter  = iw * ih;
        float ha     = A.w - A.y;
        float area_a = (A.z - A.x) * ha;
        float iou = inter / (area_a + area_b - inter + 1e-8f);
        float hr  = fminf(ha, hb) / fmaxf(fmaxf(ha, hb), 1e-8f);
        iou *= hr;

        float c0 = accN[i], c1 = accA[i], c2 = iou;
        if (!(iou > 0.f)) { c0 = 0.f; c1 = 0.f; c2 = 0.f; }

        float h1[16];
        #pragma unroll
        for (int j = 0; j < 16; ++j)
            h1[j] = fmaxf(c0 * sW1[j] + c1 * sW1[16 + j] + c2 * sW1[32 + j] + sb1[j], 0.f);
        float out = sb3;
        #pragma unroll
        for (int j2 = 0; j2 < 16; ++j2) {
            float acc = sb2[j2];
            #pragma unroll
            for (int j = 0; j < 16; ++j) acc += h1[j] * sW2[j * 16 + j2];
            out += fmaxf(acc, 0.f) * sW3[j2];
        }
        corr[((size_t)g * M_ + m) * N_ + n] = out;
    }
}

// ---------------------------------------------------------------------------
// Kernel 3: Sinkhorn. Entire 257x257 log-coupling matrix lives in LDS
// (264 KB of the WGP's 320 KB). One workgroup (32 waves) per group g.
// Wave-per-column/row, lane-strided partials, shfl_xor butterflies.
// Stride 257 floats => column access is bank-conflict free (257 % 64 == 1).
// ---------------------------------------------------------------------------
__global__ __launch_bounds__(1024, 1)
void sinkhorn_kernel(const float* __restrict__ corr,
                     const float* __restrict__ alpha,
                     const float* __restrict__ eps,
                     float* __restrict__ out)
{
    extern __shared__ float lp[];   // 257*257 floats
    const int MP = M_ + 1;          // 257
    int g   = blockIdx.x;
    int tid = threadIdx.x;
    float a0 = alpha[0];
    float inv_lam = 1.0f / (__expf(eps[0]) + 0.03f);

    for (int idx = tid; idx < MP * MP; idx += 1024) {
        int m = idx / MP, n = idx - m * MP;
        float c = (m < M_ && n < N_) ? corr[((size_t)g * M_ + m) * N_ + n] : a0;
        lp[idx] = -(1.0f - c) * inv_lam;
    }
    __syncthreads();

    const float LOG256 = 5.545177444479562f; // log(256) = log_a[M] = log_b[N]
    int wave = tid >> 5, lane = tid & 31;

    for (int it = 0; it < 9; ++it) {
        // column pass (LSE over rows); subtract (lse - log_b[col])
        for (int c = wave; c < MP; c += 32) {
            float mx = -3.0e38f;
            for (int r = lane; r < MP; r += 32) mx = fmaxf(mx, lp[r * MP + c]);
            #pragma unroll
            for (int off = 16; off; off >>= 1) mx = fmaxf(mx, __shfl_xor(mx, off, 32));
            float s = 0.f;
            for (int r = lane; r < MP; r += 32) s += __expf(lp[r * MP + c] - mx);
            #pragma unroll
            for (int off = 16; off; off >>= 1) s += __shfl_xor(s, off, 32);
            float adj = mx + __logf(s) - ((c < N_) ? 0.f : LOG256);
            for (int r = lane; r < MP; r += 32) lp[r * MP + c] -= adj;
        }
        __syncthreads();
        if (it == 8) break;  // final extra column normalization only
        // row pass (LSE over cols); subtract (lse - log_a[row])
        for (int r = wave; r < MP; r += 32) {
            float mx = -3.0e38f;
            for (int c = lane; c < MP; c += 32) mx = fmaxf(mx, lp[r * MP + c]);
            #pragma unroll
            for (int off = 16; off; off >>= 1) mx = fmaxf(mx, __shfl_xor(mx, off, 32));
            float s = 0.f;
            for (int c = lane; c < MP; c += 32) s += __expf(lp[r * MP + c] - mx);
            #pragma unroll
            for (int off = 16; off; off >>= 1) s += __shfl_xor(s, off, 32);
            float adj = mx + __logf(s) - ((r < M_) ? 0.f : LOG256);
            for (int c = lane; c < MP; c += 32) lp[r * MP + c] -= adj;
        }
        __syncthreads();
    }

    for (int idx = tid; idx < M_ * N_; idx += 1024) {
        int m = idx >> 8, n = idx & 255;
        out[((size_t)g * M_ + m) * N_ + n] = __expf(lp[m * MP + n]);
    }
}

// ---------------------------------------------------------------------------
extern "C" void kernel_launch(void* const* d_in, const int* in_sizes, int n_in,
                              void* d_out, int out_size, void* d_ws, size_t ws_size,
                              hipStream_t stream) {
    const float* tra_node = (const float*)d_in[0];
    const float* det_node = (const float*)d_in[1];
    const float* tra_app  = (const float*)d_in[2];
    const float* det_app  = (const float*)d_in[3];
    const float* tra_xyxy = (const float*)d_in[4];
    const float* det_xyxy = (const float*)d_in[5];
    const float* W1 = (const float*)d_in[6];
    const float* b1 = (const float*)d_in[7];
    const float* W2 = (const float*)d_in[8];
    const float* b2 = (const float*)d_in[9];
    const float* W3 = (const float*)d_in[10];
    const float* b3 = (const float*)d_in[11];
    const float* alpha = (const float*)d_in[12];
    const float* eps   = (const float*)d_in[13];
    float* out = (float*)d_out;

    float* ws = (float*)d_ws;
    float* traTn = ws;                                    // G*200*256
    float* detTn = traTn + (size_t)G_ * DNP_ * M_;        // G*200*256
    float* traTa = detTn + (size_t)G_ * DNP_ * N_;        // G*32*256
    float* detTa = traTa + (size_t)G_ * DA_ * M_;         // G*32*256
    float* corr  = detTa + (size_t)G_ * DA_ * N_;         // G*256*256

    // normalize + transpose: one wave per row, 4 waves per block
    {
        int waves = G_ * M_;               // 16384
        dim3 grid(waves / 4), block(128);
        normalize_t_kernel<<<grid, block, 0, stream>>>(tra_node, traTn, M_, DN_, DNP_);
        normalize_t_kernel<<<grid, block, 0, stream>>>(det_node, detTn, N_, DN_, DNP_);
        normalize_t_kernel<<<grid, block, 0, stream>>>(tra_app,  traTa, M_, DA_, DA_);
        normalize_t_kernel<<<grid, block, 0, stream>>>(det_app,  detTa, N_, DA_, DA_);
    }
    // correlation: grid (mtile=16, ntile-group=2, g=64), 8 waves/block
    {
        dim3 grid(M_ / 16, (N_ / 16) / 8, G_), block(256);
        corr_kernel<<<grid, block, 0, stream>>>(traTn, detTn, traTa, detTa,
                                                tra_xyxy, det_xyxy,
                                                W1, b1, W2, b2, W3, b3, corr);
    }
    // Sinkhorn: one workgroup per group, 257x257 matrix resident in LDS
    {
        size_t lds_bytes = (size_t)(M_ + 1) * (N_ + 1) * sizeof(float);
        sinkhorn_kernel<<<dim3(G_), dim3(1024), lds_bytes, stream>>>(corr, alpha, eps, out);
    }
}